// GNN_TRIANGLES_9328668966928
// MI455X (gfx1250) — compile-verified
//
#include <hip/hip_runtime.h>
#include <hip/hip_bf16.h>

typedef __attribute__((ext_vector_type(2))) float v2f;
typedef __attribute__((ext_vector_type(8))) float v8f;

#define HDIM 75   // hidden size
#define HP   80   // padded to multiple of 4 (K) -- also multiple of 16 (N)
#define H3   225  // 3*H
#define H3P  240  // padded
#define H2   150  // 2*H
#define H2P  160  // padded

// ---------------------------------------------------------------------------
// Elementwise helpers
// ---------------------------------------------------------------------------
__global__ void fill_zero_k(float* __restrict__ p, int n) {
  for (int i = blockIdx.x * blockDim.x + threadIdx.x; i < n;
       i += gridDim.x * blockDim.x)
    p[i] = 0.0f;
}

// S[n*80+f] = f<75 ? x[n*75+f] : 0
__global__ void init_state_k(float* __restrict__ S, const float* __restrict__ x,
                             int N) {
  int total = N * HP;
  for (int i = blockIdx.x * blockDim.x + threadIdx.x; i < total;
       i += gridDim.x * blockDim.x) {
    int node = i / HP, f = i - node * HP;
    S[i] = (f < HDIM) ? x[(size_t)node * HDIM + f] : 0.0f;
  }
}

// zero CAT pad columns 225..239
__global__ void zero_cat_pad_k(float* __restrict__ CAT, int N) {
  int total = N * (H3P - H3);
  for (int i = blockIdx.x * blockDim.x + threadIdx.x; i < total;
       i += gridDim.x * blockDim.x) {
    int node = i / (H3P - H3), j = i - node * (H3P - H3);
    CAT[(size_t)node * H3P + H3 + j] = 0.0f;
  }
}

// AGG = (1+eps)*S + AGG   (over padded N*80; pads stay 0)
__global__ void zmix_k(float* __restrict__ AGG, const float* __restrict__ S,
                       const float* __restrict__ eps, int n) {
  float sc = 1.0f + eps[0];
  for (int i = blockIdx.x * blockDim.x + threadIdx.x; i < n;
       i += gridDim.x * blockDim.x)
    AGG[i] = sc * S[i] + AGG[i];
}

// d_out[n*75+f] = S[n*80+f]
__global__ void copy_out_k(float* __restrict__ out, const float* __restrict__ S,
                           int N) {
  int total = N * HDIM;
  for (int i = blockIdx.x * blockDim.x + threadIdx.x; i < total;
       i += gridDim.x * blockDim.x) {
    int node = i / HDIM, f = i - node * HDIM;
    out[i] = S[(size_t)node * HP + f];
  }
}

// ---------------------------------------------------------------------------
// Weight packing (zero-padded; optional transpose for w_ih / w_hh)
// ---------------------------------------------------------------------------
__global__ void pack_pad_k(float* __restrict__ dst, const float* __restrict__ src,
                           int rows, int cols, int Rp, int Cp) {
  int total = Rp * Cp;
  for (int i = blockIdx.x * blockDim.x + threadIdx.x; i < total;
       i += gridDim.x * blockDim.x) {
    int r = i / Cp, c = i - r * Cp;
    dst[i] = (r < rows && c < cols) ? src[(size_t)r * cols + c] : 0.0f;
  }
}

// dst[r*Cp+c] = src[c*rows + r]  (src is (cols x rows) row-major, i.e. W^T pack)
__global__ void pack_pad_T_k(float* __restrict__ dst, const float* __restrict__ src,
                             int rows, int cols, int Rp, int Cp) {
  int total = Rp * Cp;
  for (int i = blockIdx.x * blockDim.x + threadIdx.x; i < total;
       i += gridDim.x * blockDim.x) {
    int r = i / Cp, c = i - r * Cp;
    dst[i] = (r < rows && c < cols) ? src[(size_t)c * rows + r] : 0.0f;
  }
}

// ---------------------------------------------------------------------------
// Edge scatter-add: one wave per edge, lanes stride features
// ---------------------------------------------------------------------------
__global__ void scatter_add_k(const float* __restrict__ S,
                              const int* __restrict__ srcv,
                              const int* __restrict__ dstv,
                              float* __restrict__ AGG, int E) {
  int wid = (blockIdx.x * blockDim.x + threadIdx.x) >> 5;
  int lane = threadIdx.x & 31;
  if (wid >= E) return;
  int s = srcv[wid];
  int d = dstv[wid];
  const float* xs = S + (size_t)s * HP;
  float* ad = AGG + (size_t)d * HP;
  for (int f = lane; f < HDIM; f += 32) atomicAdd(ad + f, xs[f]);
}

// ---------------------------------------------------------------------------
// Fused GRU gates + CAT slice write
// gi/gh rows are 240 wide: [r | z | n | pad]; S rows 80 wide
// ---------------------------------------------------------------------------
__global__ void gru_cat_k(const float* __restrict__ GI,
                          const float* __restrict__ GH,
                          float* __restrict__ S, float* __restrict__ CAT,
                          int it, int N) {
  int total = N * HDIM;
  for (int i = blockIdx.x * blockDim.x + threadIdx.x; i < total;
       i += gridDim.x * blockDim.x) {
    int node = i / HDIM, f = i - node * HDIM;
    const float* gi = GI + (size_t)node * H3P;
    const float* gh = GH + (size_t)node * H3P;
    float r = 1.0f / (1.0f + expf(-(gi[f] + gh[f])));
    float z = 1.0f / (1.0f + expf(-(gi[HDIM + f] + gh[HDIM + f])));
    float nn = tanhf(gi[2 * HDIM + f] + r * gh[2 * HDIM + f]);
    float h = S[(size_t)node * HP + f];
    float out = (1.0f - z) * nn + z * h;
    S[(size_t)node * HP + f] = out;
    CAT[(size_t)node * H3P + it * HDIM + f] = out;
  }
}

// ---------------------------------------------------------------------------
// fp32 WMMA GEMM:  C[16-row tile] = A[N x (KSTEPS*4)] @ B[(KSTEPS*4) x (NTILES*16)]
// One wave (32 lanes) per 16-row tile. A fragments preloaded in registers.
// A-frag layout (V_WMMA_F32_16X16X4_F32): lanes 0-15 rows M=0..15 hold K{0,1},
// lanes 16-31 same rows hold K{2,3}.  B mirrors: v0 = rows k0/k0+2, v1 = +1.
// C layout: VGPR r -> row (r + 8*half), col = lane&15.
// ---------------------------------------------------------------------------
template <int KSTEPS, int NTILES>
__launch_bounds__(32) __global__
void gemm_wmma_k(const float* __restrict__ A, int lda,
                 const float* __restrict__ B,           // ldb = NTILES*16
                 const float* __restrict__ bias, int biasCols,
                 float* __restrict__ C, int ldc, int storeCols,
                 int relu, int Nrows) {
  const int ldb = NTILES * 16;
  int tile = blockIdx.x;
  int lane = threadIdx.x & 31;
  int half = lane >> 4;
  int l15 = lane & 15;

  int row = tile * 16 + l15;
  int rowc = (row < Nrows) ? row : (Nrows - 1);  // clamp (N%16==0 normally)

  // preload A fragments
  v2f afr[KSTEPS];
  const float* arow = A + (size_t)rowc * lda + 2 * half;
#pragma unroll
  for (int ks = 0; ks < KSTEPS; ++ks)
    afr[ks] = *(const v2f*)(arow + ks * 4);

  for (int nt = 0; nt < NTILES; ++nt) {
    int col0 = nt * 16;
    v8f acc = {};
#pragma unroll
    for (int ks = 0; ks < KSTEPS; ++ks) {
      int k = ks * 4 + 2 * half;
      const float* bp = B + (size_t)k * ldb + col0 + l15;
      v2f bfr;
      bfr.x = bp[0];
      bfr.y = bp[ldb];
      acc = __builtin_amdgcn_wmma_f32_16x16x4_f32(
          false, afr[ks], false, bfr, (short)0, acc, false, false);
    }
    int col = col0 + l15;
    if (col < storeCols) {
      float bv = (col < biasCols) ? bias[col] : 0.0f;
#pragma unroll
      for (int r = 0; r < 8; ++r) {
        int rr = tile * 16 + r + half * 8;
        if (rr < Nrows) {
          float v = acc[r] + bv;
          if (relu) v = fmaxf(v, 0.0f);
          C[(size_t)rr * ldc + col] = v;
        }
      }
    }
  }
}

// ---------------------------------------------------------------------------
extern "C" void kernel_launch(void* const* d_in, const int* in_sizes, int n_in,
                              void* d_out, int out_size, void* d_ws,
                              size_t ws_size, hipStream_t stream) {
  const float* x    = (const float*)d_in[0];
  const int*   ei   = (const int*)d_in[1];
  const float* eps  = (const float*)d_in[2];
  const float* w1   = (const float*)d_in[3];
  const float* b1   = (const float*)d_in[4];
  const float* w2   = (const float*)d_in[5];
  const float* b2   = (const float*)d_in[6];
  const float* w_ih = (const float*)d_in[7];
  const float* w_hh = (const float*)d_in[8];
  const float* b_ih = (const float*)d_in[9];
  const float* b_hh = (const float*)d_in[10];
  const float* wc1  = (const float*)d_in[11];
  const float* bc1  = (const float*)d_in[12];
  const float* wc2  = (const float*)d_in[13];
  const float* bc2  = (const float*)d_in[14];

  const int N = in_sizes[0] / HDIM;
  const int E = in_sizes[1] / 2;
  const int* srcv = ei;
  const int* dstv = ei + E;

  // workspace layout (floats)
  float* S   = (float*)d_ws;                 // [N x 80]  node state (xt == h)
  float* BUF = S + (size_t)N * HP;           // [N x 240] AGG/T1/M, reused as GH
  float* GI  = BUF + (size_t)N * H3P;        // [N x 240] GRU input gates; reused as R1
  float* CAT = GI + (size_t)N * H3P;         // [N x 240] concat of 3 states
  float* W1p  = CAT + (size_t)N * H3P;       // 80x80
  float* W2p  = W1p + HP * HP;               // 80x80
  float* WIHp = W2p + HP * HP;               // 80x240 (w_ih^T padded)
  float* WHHp = WIHp + HP * H3P;             // 80x240 (w_hh^T padded)
  float* WC1p = WHHp + HP * H3P;             // 240x160
  float* WC2p = WC1p + H3P * H2P;            // 160x80
  float* AGGZ = BUF;                         // [N x 80]
  float* T1   = BUF + (size_t)N * HP;        // [N x 80]
  float* M    = BUF + (size_t)N * 2 * HP;    // [N x 80]
  float* GH   = BUF;                         // [N x 240] (after M consumed)
  float* R1   = GI;                          // [N x 160] readout hidden

  float* outS = (float*)d_out;               // xt, N*75
  float* outR = outS + (size_t)N * HDIM;     // readout, N*75

  const int TPB = 256;
  auto blk = [](int n, int t) { return (n + t - 1) / t; };
  const int tiles = (N + 15) / 16;           // 6250 for N=100000

  // ---- pack weights (every call; deterministic, cheap) ----
  pack_pad_k<<<blk(HP * HP, TPB), TPB, 0, stream>>>(W1p, w1, HDIM, HDIM, HP, HP);
  pack_pad_k<<<blk(HP * HP, TPB), TPB, 0, stream>>>(W2p, w2, HDIM, HDIM, HP, HP);
  pack_pad_T_k<<<blk(HP * H3P, TPB), TPB, 0, stream>>>(WIHp, w_ih, HDIM, H3, HP, H3P);
  pack_pad_T_k<<<blk(HP * H3P, TPB), TPB, 0, stream>>>(WHHp, w_hh, HDIM, H3, HP, H3P);
  pack_pad_k<<<blk(H3P * H2P, TPB), TPB, 0, stream>>>(WC1p, wc1, H3, H2, H3P, H2P);
  pack_pad_k<<<blk(H2P * HP, TPB), TPB, 0, stream>>>(WC2p, wc2, H2, HDIM, H2P, HP);

  // ---- init state & CAT pads ----
  init_state_k<<<blk(N * HP, TPB), TPB, 0, stream>>>(S, x, N);
  zero_cat_pad_k<<<blk(N * (H3P - H3), TPB), TPB, 0, stream>>>(CAT, N);

  // ---- 3 GIN + GRU iterations ----
  for (int it = 0; it < 3; ++it) {
    fill_zero_k<<<blk(N * HP, TPB), TPB, 0, stream>>>(AGGZ, N * HP);
    scatter_add_k<<<blk(E * 32, TPB), TPB, 0, stream>>>(S, srcv, dstv, AGGZ, E);
    zmix_k<<<blk(N * HP, TPB), TPB, 0, stream>>>(AGGZ, S, eps, N * HP);
    // T1 = relu(Z @ w1 + b1)
    gemm_wmma_k<20, 5><<<tiles, 32, 0, stream>>>(AGGZ, HP, W1p, b1, HDIM,
                                                 T1, HP, HP, 1, N);
    // M = relu(T1 @ w2 + b2)
    gemm_wmma_k<20, 5><<<tiles, 32, 0, stream>>>(T1, HP, W2p, b2, HDIM,
                                                 M, HP, HP, 1, N);
    // GI = M @ w_ih^T + b_ih
    gemm_wmma_k<20, 15><<<tiles, 32, 0, stream>>>(M, HP, WIHp, b_ih, H3,
                                                  GI, H3P, H3P, 0, N);
    // GH = S @ w_hh^T + b_hh   (overwrites BUF; M already consumed)
    gemm_wmma_k<20, 15><<<tiles, 32, 0, stream>>>(S, HP, WHHp, b_hh, H3,
                                                  GH, H3P, H3P, 0, N);
    // S = GRU(GI, GH, S); CAT[:, it*75:(it+1)*75] = S
    gru_cat_k<<<blk(N * HDIM, TPB), TPB, 0, stream>>>(GI, GH, S, CAT, it, N);
  }

  // ---- readout ----
  // R1 = relu(CAT @ wc1 + bc1)
  gemm_wmma_k<60, 10><<<tiles, 32, 0, stream>>>(CAT, H3P, WC1p, bc1, H2,
                                                R1, H2P, H2P, 1, N);
  // out = R1 @ wc2 + bc2  (unpadded store directly into d_out)
  gemm_wmma_k<40, 5><<<tiles, 32, 0, stream>>>(R1, H2P, WC2p, bc2, HDIM,
                                               outR, HDIM, HDIM, 0, N);
  // xt -> d_out
  copy_out_k<<<blk(N * HDIM, TPB), TPB, 0, stream>>>(outS, S, N);
}